// attention_5214090297751
// MI455X (gfx1250) — compile-verified
//
#include <hip/hip_runtime.h>

typedef __attribute__((ext_vector_type(2))) float v2f;
typedef __attribute__((ext_vector_type(4))) float v4f;
typedef __attribute__((ext_vector_type(8))) float v8f;

#define B_   32
#define DE_  256
#define T_   32
#define C_   128
#define HW_  (128 * 128)

#define ET_S 132   // eT row stride (padded): [t][c]
#define EC_S 34    // eC row stride (padded): [c][t]
#define H_S  132   // h tile row stride:      [n][c]
#define BT_S 34    // beta tile row stride:   [n][t]

// ---------------------------------------------------------------------------
// Kernel 1: e_[b,c,t] = sum_d Wd[d,c] * e[b,d,t] + bias[c]   (tiny: 67 MFLOP)
// grid = B*16 blocks x 256 threads; one output per thread.
// Regular (RT) stores: e_ is re-read by all 4096 blocks of kernel 2 and we
// WANT it resident in L2.
// ---------------------------------------------------------------------------
__global__ __launch_bounds__(256) void dense_e_kernel(
    const float* __restrict__ e, const float* __restrict__ Wd,
    const float* __restrict__ bias, float* __restrict__ e_out)
{
    const int blk = blockIdx.x;                 // 512 blocks
    const int b   = blk >> 4;
    const int o   = ((blk & 15) << 8) + threadIdx.x;   // 0..4095 = c*32+t
    const int c   = o >> 5;
    const int t   = o & 31;
    const float* ep = e + (size_t)b * DE_ * T_ + t;    // stride T_ over d
    const float* wp = Wd + c;                          // stride C_ over d
    float acc = bias[c];
    #pragma unroll 8
    for (int d = 0; d < DE_; ++d)
        acc = fmaf(wp[d * C_], ep[d * T_], acc);
    e_out[(size_t)b * (C_ * T_) + o] = acc;
}

// ---------------------------------------------------------------------------
// Kernel 2: fused scores -> softmax(T) -> context -> concat output.
// grid = B * (HW/128) blocks x 256 threads (8 waves, 16 pixels per wave).
// h: streamed once (NT loads). out: streamed once, never re-read (NT stores).
// ---------------------------------------------------------------------------
__global__ __launch_bounds__(256) void attn_main_kernel(
    const float* __restrict__ h, const float* __restrict__ e_,
    float* __restrict__ out)
{
    __shared__ float eT[T_ * ET_S];       // e_^T   [t][c]   16896 B
    __shared__ float eC[C_ * EC_S];       // e_     [c][t]   17408 B
    __shared__ float hS[8][16 * H_S];     // h tile [n][c]   67584 B (per wave)
    __shared__ float bS[8][16 * BT_S];    // beta   [n][t]   17408 B (per wave)

    const int b    = blockIdx.x >> 7;
    const int tile = blockIdx.x & 127;
    const int tid  = threadIdx.x;
    const int wave = tid >> 5;
    const int lane = tid & 31;
    const int nl   = lane & 15;
    const int half = lane >> 4;

    // ---- stage e_ for this batch into both LDS layouts (reads hit L2) ----
    const float* eb = e_ + (size_t)b * (C_ * T_);
    for (int i = tid; i < C_ * T_; i += 256) {
        float v = eb[i];
        int c = i >> 5, t = i & 31;
        eT[t * ET_S + c] = v;
        eC[c * EC_S + t] = v;
    }

    // ---- stream h tile once: -> out[...,128:256] and -> LDS ----
    const int n0 = tile * 128 + wave * 16;   // this wave's global pixel base
    const v4f* hsrc = (const v4f*)(h + ((size_t)b * HW_ + n0) * C_);
    float* outb = out + ((size_t)b * HW_ + n0) * (2 * C_);
    #pragma unroll
    for (int j = 0; j < 16; ++j) {
        int fi  = j * 32 + lane;             // float4 index within 2048 floats
        v4f v = __builtin_nontemporal_load(hsrc + fi);
        int off = fi * 4;
        int p   = off >> 7;                  // pixel 0..15 within tile
        int c   = off & 127;                 // channel (multiple of 4)
        __builtin_nontemporal_store(v, (v4f*)(outb + (size_t)p * (2 * C_) + C_ + c));
        *(v4f*)(&hS[wave][p * H_S + c]) = v;
    }
    __syncthreads();

    // ---- scores: S[t,n] = sum_c e_[c,t] * h[c,n], two 16x16 t-tiles ----
    v8f s0 = {0, 0, 0, 0, 0, 0, 0, 0};
    v8f s1 = {0, 0, 0, 0, 0, 0, 0, 0};
    const float* hrow = &hS[wave][nl * H_S + 2 * half];
    const float* eA0  = &eT[nl * ET_S + 2 * half];
    const float* eA1  = &eT[(16 + nl) * ET_S + 2 * half];
    #pragma unroll
    for (int kk = 0; kk < 32; ++kk) {
        int kb = kk * 4;
        v2f bv = *(const v2f*)(hrow + kb);   // B[k][n], k = kb+2*half, +1
        v2f a0 = *(const v2f*)(eA0 + kb);    // A[t][k] for t-tile 0
        v2f a1 = *(const v2f*)(eA1 + kb);    // A[t][k] for t-tile 1
        s0 = __builtin_amdgcn_wmma_f32_16x16x4_f32(false, a0, false, bv,
                                                   (short)0, s0, false, false);
        s1 = __builtin_amdgcn_wmma_f32_16x16x4_f32(false, a1, false, bv,
                                                   (short)0, s1, false, false);
    }

    // ---- softmax over T=32 (regs r -> t = r + 8*half (+16 for s1)) ----
    float m = s0[0];
    #pragma unroll
    for (int r = 0; r < 8; ++r) {
        m = fmaxf(m, s0[r]);
        m = fmaxf(m, s1[r]);
    }
    m = fmaxf(m, __shfl_xor(m, 16, 32));     // combine with partner lane
    float p0[8], p1[8];
    float sum = 0.f;
    #pragma unroll
    for (int r = 0; r < 8; ++r) {
        p0[r] = __expf(s0[r] - m); sum += p0[r];
        p1[r] = __expf(s1[r] - m); sum += p1[r];
    }
    sum += __shfl_xor(sum, 16, 32);
    float inv = 1.0f / sum;
    #pragma unroll
    for (int r = 0; r < 8; ++r) {
        bS[wave][nl * BT_S + (r + 8 * half)]        = p0[r] * inv;
        bS[wave][nl * BT_S + 16 + (r + 8 * half)]   = p1[r] * inv;
    }
    // same-wave DS ops are in-order: subsequent reads see these stores.

    // preload beta B-operands (shared by all 8 m-tiles)
    v2f bb[8];
    #pragma unroll
    for (int kk = 0; kk < 8; ++kk)
        bb[kk] = *(const v2f*)(&bS[wave][nl * BT_S + kk * 4 + 2 * half]);

    // ---- context: C[c,n] = sum_t e_[c,t] * beta[t,n]; write out[...,0:128] ----
    #pragma unroll
    for (int mt = 0; mt < 8; ++mt) {
        int cb = mt * 16;
        v8f acc = {0, 0, 0, 0, 0, 0, 0, 0};
        const float* ea = &eC[(cb + nl) * EC_S + 2 * half];
        #pragma unroll
        for (int kk = 0; kk < 8; ++kk) {
            v2f a = *(const v2f*)(ea + kk * 4);     // A[c][t] pair
            acc = __builtin_amdgcn_wmma_f32_16x16x4_f32(false, a, false, bb[kk],
                                                        (short)0, acc, false, false);
        }
        // acc[r] -> channel cb + 8*half + r, pixel nl
        float* row = outb + (size_t)nl * (2 * C_) + cb + 8 * half;
        v4f lo = {acc[0], acc[1], acc[2], acc[3]};
        v4f hi = {acc[4], acc[5], acc[6], acc[7]};
        __builtin_nontemporal_store(lo, (v4f*)(row));
        __builtin_nontemporal_store(hi, (v4f*)(row + 4));
    }
}

// ---------------------------------------------------------------------------
extern "C" void kernel_launch(void* const* d_in, const int* in_sizes, int n_in,
                              void* d_out, int out_size, void* d_ws, size_t ws_size,
                              hipStream_t stream)
{
    const float* e    = (const float*)d_in[0];   // [B, DE, T]
    const float* h    = (const float*)d_in[1];   // [B, H, W, C]
    const float* Wd   = (const float*)d_in[2];   // [DE, C]
    const float* bias = (const float*)d_in[3];   // [C]
    float* out = (float*)d_out;                  // [B, H, W, 2C]
    float* e_  = (float*)d_ws;                   // B*C*T floats = 512 KB scratch

    dense_e_kernel<<<B_ * 16, 256, 0, stream>>>(e, Wd, bias, e_);
    attn_main_kernel<<<B_ * (HW_ / 128), 256, 0, stream>>>(h, e_, out);
}